// MoE_78726750536466
// MI455X (gfx1250) — compile-verified
//
#include <hip/hip_runtime.h>
#include <cstddef>

typedef __attribute__((ext_vector_type(2))) float v2f;
typedef __attribute__((ext_vector_type(8))) float v8f;

namespace {
constexpr int NE = 8, TOPK = 2, CH = 256, NGATE = 4;
constexpr int NB = 8, HH = 32, WW = 32;
constexpr int HW = HH * WW;                 // 1024 pixels
constexpr int TH = 8, TW = 8, TPIX = 64;    // spatial tile
constexpr int CP = CH / 2;                  // 128 channel pairs
constexpr int XS_SZ = CH * 100;             // halo patch, pair-interleaved [cp][100][2]
constexpr int US_SZ = CH * TPIX;            // u tile, pair-interleaved [cp][64][2]
constexpr int LDS_FLOATS = XS_SZ + US_SZ + TPIX + 256;
}

static __device__ __forceinline__ v8f wmma4(v2f a, v2f b, v8f c) {
  // D = A(16x4,f32) * B(4x16,f32) + C(16x16,f32)
  return __builtin_amdgcn_wmma_f32_16x16x4_f32(
      /*neg_a=*/false, a, /*neg_b=*/false, b,
      /*c_mod=*/(short)0, c, /*reuse_a=*/false, /*reuse_b=*/false);
}

// ---- conv weight pre-transpose, pair-interleaved for b64 A-fragment loads:
// wt[((e*9+tap)*CP + cp)*(CH*2) + o*2 + j] = conv_w[((e*CH+o)*CH + 2cp+j)*9 + tap]
__global__ void tconv_kernel(const float* __restrict__ cw, float* __restrict__ wt) {
  const int idx = blockIdx.x;              // (e*9 + tap)*CP + cp
  const int o = threadIdx.x;
  const int cp = idx & (CP - 1);
  const int et = idx / CP;
  const int tap = et % 9;
  const int e = et / 9;
  const size_t src = ((size_t)(e * CH + o) * CH + 2 * cp) * 9 + tap;
  float* dst = wt + (size_t)idx * (CH * 2) + o * 2;
  dst[0] = cw[src];
  dst[1] = cw[src + 9];
}
// pt[(e*CP + cp)*(CH*2) + o*2 + j] = proj_w[(e*CH+o)*CH + 2cp+j]
__global__ void tproj_kernel(const float* __restrict__ pw, float* __restrict__ pt) {
  const int idx = blockIdx.x;              // e*CP + cp
  const int o = threadIdx.x;
  const int cp = idx & (CP - 1);
  const int e = idx / CP;
  const size_t src = (size_t)(e * CH + o) * CH + 2 * cp;
  float* dst = pt + (size_t)idx * (CH * 2) + o * 2;
  dst[0] = pw[src];
  dst[1] = pw[src + 1];
}

// ---- global average pool: xgap[b*CH+c] = mean over 1024 pixels
__global__ void gap_kernel(const float* __restrict__ x, float* __restrict__ xgap) {
  __shared__ float red[256];
  const int bc = blockIdx.x;
  const float* p = x + (size_t)bc * HW;
  float s = 0.0f;
  for (int i = threadIdx.x; i < HW; i += 256) s += p[i];
  red[threadIdx.x] = s;
  __syncthreads();
  for (int st = 128; st > 0; st >>= 1) {
    if (threadIdx.x < st) red[threadIdx.x] += red[threadIdx.x + st];
    __syncthreads();
  }
  if (threadIdx.x == 0) xgap[bc] = red[0] * (1.0f / (float)HW);
}

// ---- gating: softmax, top-2, renorm weights, balance loss
__global__ void gate_kernel(const float* __restrict__ xgap,
                            const float* __restrict__ gates,
                            int* __restrict__ ti, float* __restrict__ tw,
                            float* __restrict__ loss_out) {
  __shared__ float probs_s[NGATE * NB * NE];
  __shared__ float bal[NGATE];
  const int t = threadIdx.x;
  if (t < NGATE * NB) {
    const int gi = t >> 3, b = t & 7;
    const float* g = gates + (size_t)gi * CH * NE;
    float logit[NE];
#pragma unroll
    for (int e = 0; e < NE; ++e) logit[e] = 0.0f;
    for (int c = 0; c < CH; ++c) {
      const float xv = xgap[b * CH + c];
#pragma unroll
      for (int e = 0; e < NE; ++e) logit[e] += xv * g[c * NE + e];
    }
    float mx = logit[0];
#pragma unroll
    for (int e = 1; e < NE; ++e) mx = fmaxf(mx, logit[e]);
    float pr[NE]; float sum = 0.0f;
#pragma unroll
    for (int e = 0; e < NE; ++e) { pr[e] = expf(logit[e] - mx); sum += pr[e]; }
    const float inv = 1.0f / sum;
#pragma unroll
    for (int e = 0; e < NE; ++e) { pr[e] *= inv; probs_s[t * NE + e] = pr[e]; }
    int e0 = 0;
#pragma unroll
    for (int e = 1; e < NE; ++e) if (pr[e] > pr[e0]) e0 = e;
    int e1 = (e0 == 0) ? 1 : 0;
#pragma unroll
    for (int e = 0; e < NE; ++e) if (e != e0 && pr[e] > pr[e1]) e1 = e;
    const float p0 = pr[e0], p1 = pr[e1];
    const float m2 = fmaxf(p0, p1);
    const float w0 = expf(p0 - m2), w1 = expf(p1 - m2);
    const float wsum = w0 + w1;
    ti[t * TOPK + 0] = e0; ti[t * TOPK + 1] = e1;
    tw[t * TOPK + 0] = w0 / wsum; tw[t * TOPK + 1] = w1 / wsum;
  }
  __syncthreads();
  if (t < NGATE) {
    float v[NE]; float mean = 0.0f;
#pragma unroll
    for (int e = 0; e < NE; ++e) {
      float s = 0.0f;
      for (int b = 0; b < NB; ++b) s += probs_s[(t * NB + b) * NE + e];
      v[e] = s; mean += s;
    }
    mean *= (1.0f / (float)NE);
    float var = 0.0f;
#pragma unroll
    for (int e = 0; e < NE; ++e) { const float d = v[e] - mean; var += d * d; }
    var *= (1.0f / (float)NE);
    bal[t] = var / (mean * mean + 1e-10f);
  }
  __syncthreads();
  if (t == 0) loss_out[0] = (bal[0] + bal[1] + bal[2] + bal[3]) * 0.25f;
}

// ---- fused expert: conv3x3 (9 shifted GEMMs) + channel norm + proj, f32 WMMA
__global__ __launch_bounds__(256, 1) void moe_expert_kernel(
    const float* __restrict__ x, const float* __restrict__ wt,
    const float* __restrict__ cb, const float* __restrict__ pt,
    const float* __restrict__ pb, const int* __restrict__ ti,
    const float* __restrict__ tw, float* __restrict__ out) {
  extern __shared__ float smem[];
  float* xs = smem;                       // [CP][100][2] halo patch
  float* us = smem + XS_SZ;               // [CP][64][2] u tile
  float* sc = us + US_SZ;                 // [64] per-pixel scale
  float* part = sc + TPIX;                // [256] partial sums

  const int tid = threadIdx.x;
  const int lane = tid & 31;
  const int lm = lane & 15;
  const int hf = lane >> 4;               // upper half-wave: K+2 / row+8
  const int wave = tid >> 5;
  const int job = blockIdx.x >> 4;        // gi*NB + b
  const int tile = blockIdx.x & 15;
  const int gi = job >> 3;
  const int b = job & 7;
  const int ty0 = (tile >> 2) * TH;
  const int tx0 = (tile & 3) * TW;
  const int m0 = wave * 32;               // this wave's 32 output rows

  // stage halo'd input patch into LDS, pair-interleaved (zero pad == 'SAME')
  const float* xb = x + (size_t)b * CH * HW;
  for (int i = tid; i < XS_SZ; i += 256) {
    const int cp = i / 200, r = i - cp * 200;
    const int off = r >> 1, j = r & 1;
    const int c = cp * 2 + j;
    const int yy = off / 10, xx = off - yy * 10;
    const int gy = ty0 + yy - 1, gx = tx0 + xx - 1;
    float v = 0.0f;
    if (gy >= 0 && gy < HH && gx >= 0 && gx < WW) v = xb[c * HW + gy * WW + gx];
    xs[i] = v;
  }
  __syncthreads();

  v8f oacc[2][4] = {};

  for (int k = 0; k < TOPK; ++k) {
    const int e = ti[(gi * NB + b) * TOPK + k];
    const float gw = tw[(gi * NB + b) * TOPK + k];

    // ---- conv: u(256x64) += W_tap(256x256) @ x_shift(256x64), 9 taps
    v8f acc[2][4] = {};
    const float* wte = wt + (size_t)e * 9 * CP * (CH * 2);
    for (int tap = 0; tap < 9; ++tap) {
      const int kh = tap / 3, kw = tap - kh * 3;
      const float* wtt = wte + (size_t)tap * CP * (CH * 2);
      const int boff = kh * 10 + kw;
#pragma unroll 2
      for (int k0 = 0; k0 < CH; k0 += 4) {
        const int cp = (k0 >> 1) + hf;
        const float* wr = wtt + (size_t)cp * (CH * 2);
        v2f a[2], bm[4];
        a[0] = *(const v2f*)(wr + (m0 + lm) * 2);
        a[1] = *(const v2f*)(wr + (m0 + 16 + lm) * 2);
        const float* xrow = xs + (size_t)cp * 200 + boff * 2;
#pragma unroll
        for (int nt = 0; nt < 4; ++nt) {
          const int p = nt * 16 + lm;
          const int off = (p >> 3) * 10 + (p & 7);
          bm[nt] = *(const v2f*)(xrow + off * 2);
        }
#pragma unroll
        for (int mt = 0; mt < 2; ++mt)
#pragma unroll
          for (int nt = 0; nt < 4; ++nt)
            acc[mt][nt] = wmma4(a[mt], bm[nt], acc[mt][nt]);
      }
    }

    __syncthreads();  // previous iteration's proj reads of us are done
#pragma unroll
    for (int mt = 0; mt < 2; ++mt) {
      const int rowb = m0 + mt * 16 + hf * 8;  // even
#pragma unroll
      for (int nt = 0; nt < 4; ++nt) {
        const int p = nt * 16 + lm;
#pragma unroll
        for (int r = 0; r < 8; r += 2) {
          float* d = us + (size_t)((rowb + r) >> 1) * 128 + p * 2;
          d[0] = acc[mt][nt][r]     + cb[e * CH + rowb + r];
          d[1] = acc[mt][nt][r + 1] + cb[e * CH + rowb + r + 1];
        }
      }
    }
    __syncthreads();

    // ---- per-pixel norm: sn = sum_c u^2 ; scale = sn/((1+sn)(sqrt(sn)+1e-8))
    {
      const int q = tid >> 6, p = tid & 63;  // 4-way channel split
      float sn = 0.0f;
      for (int cp = q * (CP / 4); cp < (q + 1) * (CP / 4); ++cp) {
        const v2f v = *(const v2f*)(us + (size_t)cp * 128 + p * 2);
        sn += v.x * v.x + v.y * v.y;
      }
      part[tid] = sn;
    }
    __syncthreads();
    if (tid < TPIX) {
      const float s = part[tid] + part[tid + 64] + part[tid + 128] + part[tid + 192];
      sc[tid] = s / ((1.0f + s) * (sqrtf(s) + 1e-8f));
    }
    __syncthreads();
    {
      v2f* uv = (v2f*)us;
      for (int i = tid; i < US_SZ / 2; i += 256) {
        v2f v = uv[i];
        const float s = sc[i & (TPIX - 1)];
        v.x *= s; v.y *= s;
        uv[i] = v;
      }
    }
    __syncthreads();

    // ---- projection: y(256x64) = P(256x256) @ u_norm(256x64)
    v8f pacc[2][4] = {};
    const float* pte = pt + (size_t)e * CP * (CH * 2);
#pragma unroll 2
    for (int k0 = 0; k0 < CH; k0 += 4) {
      const int cp = (k0 >> 1) + hf;
      const float* pr = pte + (size_t)cp * (CH * 2);
      v2f a[2], bm[4];
      a[0] = *(const v2f*)(pr + (m0 + lm) * 2);
      a[1] = *(const v2f*)(pr + (m0 + 16 + lm) * 2);
      const float* ur = us + (size_t)cp * 128;
#pragma unroll
      for (int nt = 0; nt < 4; ++nt)
        bm[nt] = *(const v2f*)(ur + (nt * 16 + lm) * 2);
#pragma unroll
      for (int mt = 0; mt < 2; ++mt)
#pragma unroll
        for (int nt = 0; nt < 4; ++nt)
          pacc[mt][nt] = wmma4(a[mt], bm[nt], pacc[mt][nt]);
    }

    // weighted accumulate (+ proj bias) into output fragments
#pragma unroll
    for (int mt = 0; mt < 2; ++mt) {
      const int rowb = m0 + mt * 16 + hf * 8;
#pragma unroll
      for (int nt = 0; nt < 4; ++nt)
#pragma unroll
        for (int r = 0; r < 8; ++r)
          oacc[mt][nt][r] += gw * (pacc[mt][nt][r] + pb[e * CH + rowb + r]);
    }
  }

  // ---- store: out[gi][b][row][ty0+py][tx0+px]
  float* ob = out + (size_t)(gi * NB + b) * CH * HW;
#pragma unroll
  for (int mt = 0; mt < 2; ++mt) {
    const int rowb = m0 + mt * 16 + hf * 8;
#pragma unroll
    for (int nt = 0; nt < 4; ++nt) {
      const int p = nt * 16 + lm;
      const int py = p >> 3, px = p & 7;
#pragma unroll
      for (int r = 0; r < 8; ++r)
        ob[(size_t)(rowb + r) * HW + (ty0 + py) * WW + (tx0 + px)] = oacc[mt][nt][r];
    }
  }
}

extern "C" void kernel_launch(void* const* d_in, const int* in_sizes, int n_in,
                              void* d_out, int out_size, void* d_ws, size_t ws_size,
                              hipStream_t stream) {
  const float* x      = (const float*)d_in[0];
  const float* gates  = (const float*)d_in[1];
  const float* conv_w = (const float*)d_in[2];
  const float* conv_b = (const float*)d_in[3];
  const float* proj_w = (const float*)d_in[4];
  const float* proj_b = (const float*)d_in[5];
  float* out = (float*)d_out;

  // workspace layout (~21 MB): pair-interleaved transposed weights + gate scratch
  float* ws   = (float*)d_ws;
  float* wt   = ws;                                 // NE*9*CP*(CH*2)
  float* pt   = wt + (size_t)NE * 9 * CH * CH;      // NE*CP*(CH*2)
  float* xgap = pt + (size_t)NE * CH * CH;          // NB*CH
  float* twb  = xgap + NB * CH;                     // NGATE*NB*TOPK
  int*   tib  = (int*)(twb + NGATE * NB * TOPK);    // NGATE*NB*TOPK

  const size_t smem = (size_t)LDS_FLOATS * sizeof(float);  // ~169 KB
  hipFuncSetAttribute((const void*)moe_expert_kernel,
                      hipFuncAttributeMaxDynamicSharedMemorySize, (int)smem);

  tconv_kernel<<<NE * 9 * CP, 256, 0, stream>>>(conv_w, wt);
  tproj_kernel<<<NE * CP, 256, 0, stream>>>(proj_w, pt);
  gap_kernel<<<NB * CH, 256, 0, stream>>>(x, xgap);
  gate_kernel<<<1, 64, 0, stream>>>(xgap, gates, tib, twb,
                                    out + (size_t)NGATE * NB * CH * HW);
  // 32 (gate,batch) jobs x 16 spatial tiles
  moe_expert_kernel<<<NGATE * NB * 16, 256, smem, stream>>>(
      x, wt, conv_b, pt, proj_b, tib, twb, out);
}